// NNModelSPeciesWise_4174708212170
// MI455X (gfx1250) — compile-verified
//
#include <hip/hip_runtime.h>
#include <stdint.h>

// ---------------- constants (from reference) ----------------
#define NSPEC   4
#define NSTRUCT 1000
#define FDIM    256            // feature dim == hidden dim
#define WAVES_PER_BLOCK 4      // 4 waves = 128 threads; block owns 64 atoms
#define ASTRIDE 260            // padded LDS row stride (bank-conflict-free A frags)
#define PSTRIDE 17             // padded W-panel row stride (conflict-free B frags)
#define PANEL_FLOATS (FDIM * PSTRIDE)          // 256 x 16 panel, stride 17
#define PERWAVE_FLOATS (2 * 16 * ASTRIDE + 32) // two 16x256 tiles + aux

typedef __attribute__((ext_vector_type(2))) float v2f;
typedef __attribute__((ext_vector_type(8))) float v8f;

// D = A(16x4 f32) * B(4x16 f32) + C(16x16 f32)  -- exact fp32 matrix op
__device__ __forceinline__ v8f wmma4(v2f a, v2f b, v8f c) {
  return __builtin_amdgcn_wmma_f32_16x16x4_f32(
      /*neg_a=*/false, a, /*neg_b=*/false, b,
      /*c_mod=*/(short)0, c, /*reuse_a=*/false, /*reuse_b=*/false);
}

// Branchless tanh on the hw transcendental pipe:
// tanh(x) = sign(x) * (1 - 2/(exp2(2*log2e*|x|)+1)); exp2->inf saturates to +-1.
__device__ __forceinline__ float fast_tanh(float x) {
  const float ax = __builtin_fabsf(x);
  const float e  = __builtin_amdgcn_exp2f(ax * 2.88539008177793f); // 2*log2(e)
  const float r  = __builtin_amdgcn_rcpf(e + 1.0f);
  return __builtin_copysignf(1.0f - 2.0f * r, x);
}

// One 16x16 output tile: acc += A[16x256 in LDS, stride ASTRIDE] x Wpanel[256x16 in LDS]
__device__ __forceinline__ v8f gemm_tile_lds(const float* __restrict__ bufIn,
                                             const float* __restrict__ wpan,
                                             int lane, v8f acc) {
  const int m    = lane & 15;   // A row / B col within tile
  const int half = lane >> 4;   // K sub-pair select (16x16x4 A/B layout)
  #pragma unroll 4
  for (int k64 = 0; k64 < 64; ++k64) {
    const int kb = (k64 << 2) + (half << 1);
    v2f a = *reinterpret_cast<const v2f*>(&bufIn[m * ASTRIDE + kb]);
    v2f b;
    b.x = wpan[kb * PSTRIDE + m];
    b.y = wpan[(kb + 1) * PSTRIDE + m];
    acc = wmma4(a, b, acc);
  }
  return acc;
}

// ---------------- setup kernels ----------------
__global__ void k_init(float* __restrict__ out, int nOut,
                       int* __restrict__ counts, int* __restrict__ binned,
                       int nBinned) {
  int i = blockIdx.x * blockDim.x + threadIdx.x;
  int stride = gridDim.x * blockDim.x;
  for (int k = i; k < nOut; k += stride) out[k] = 0.0f;
  for (int k = i; k < nBinned; k += stride) binned[k] = -1;
  if (i < 8) counts[i] = 0;   // counts[0..3], cursors[4..7]
}

__global__ void k_count(const int* __restrict__ z, int* __restrict__ counts, int N) {
  int i = blockIdx.x * blockDim.x + threadIdx.x;
  if (i < N) atomicAdd(&counts[z[i]], 1);
}

__global__ void k_offsets(const int* __restrict__ counts, int* __restrict__ offs) {
  int off = 0;
  for (int s = 0; s < NSPEC; ++s) {
    offs[s] = off;
    off += (counts[s] + 63) & ~63;   // pad each species bin to 64 rows (1 block)
  }
  offs[NSPEC] = off;
}

__global__ void k_scatter(const int* __restrict__ z, int* __restrict__ cursors,
                          const int* __restrict__ offs, int* __restrict__ binned,
                          int N) {
  int i = blockIdx.x * blockDim.x + threadIdx.x;
  if (i < N) {
    int s = z[i];
    int pos = atomicAdd(&cursors[s], 1);
    binned[offs[s] + pos] = i;
  }
}

// ---------------- fused MLP forward + backward (block = 64 atoms, 1 species) ----------------
__global__ void k_mlp(const float* __restrict__ ps,   // [N,256]
                      const int* __restrict__ smap,   // [N]
                      const int* __restrict__ offs,   // [5]
                      const int* __restrict__ binned, // 64-padded bins
                      const float* __restrict__ W1, const float* __restrict__ b1,
                      const float* __restrict__ W2, const float* __restrict__ b2,
                      const float* __restrict__ W3, const float* __restrict__ b3,
                      const float* __restrict__ Wsk, const float* __restrict__ bsk,
                      float* __restrict__ energies,   // [NSTRUCT]
                      float* __restrict__ ngrads) {   // [N,256]
  extern __shared__ float smem[];
  const int tid  = threadIdx.x;
  const int lane = tid & 31;
  const int wave = tid >> 5;

  float* wpan = smem;                                  // shared 256x16 W panel
  float* bufA = smem + PANEL_FLOATS + wave * PERWAVE_FLOATS;
  float* bufB = bufA + 16 * ASTRIDE;
  float* auxf = bufB + 16 * ASTRIDE;                   // [16] skip dots
  int*   auxi = (int*)(auxf + 16);                     // [16] atom ids

  const int row0blk = (int)blockIdx.x << 6;            // 64 rows per block
  const int total   = offs[NSPEC];
  if (row0blk >= total) return;                        // uniform over the block

  int sp = 0;
  #pragma unroll
  for (int s = 1; s < NSPEC; ++s)
    if (row0blk >= offs[s]) sp = s;                    // block is single-species

  const int row0 = row0blk + (wave << 4);
  const int m    = lane & 15;
  const int half = lane >> 4;

  const int idxm = binned[row0 + m];
  if (half == 0) auxi[m] = idxm;

  const float* pW1 = W1 + (size_t)sp * FDIM * FDIM;
  const float* pW2 = W2 + (size_t)sp * FDIM * FDIM;
  const float* pb1 = b1 + sp * FDIM;
  const float* pb2 = b2 + sp * FDIM;
  const float* pw3 = W3 + sp * FDIM;
  const float* wsk = Wsk + sp * FDIM;

  // ---- stage X tile into bufA; fold in skip dot x . Wskip ----
  const float* xrow = ps + (size_t)(idxm < 0 ? 0 : idxm) * FDIM;
  const int fbase = half * 128;
  float sk = 0.0f;
  for (int f = 0; f < 128; f += 4) {
    float4 v = {0.f, 0.f, 0.f, 0.f};
    if (idxm >= 0) v = *reinterpret_cast<const float4*>(&xrow[fbase + f]);
    *reinterpret_cast<float4*>(&bufA[m * ASTRIDE + fbase + f]) = v;
    sk += v.x * wsk[fbase + f]     + v.y * wsk[fbase + f + 1]
        + v.z * wsk[fbase + f + 2] + v.w * wsk[fbase + f + 3];
  }
  sk += __shfl_xor(sk, 16, 32);
  if (half == 0) auxf[m] = sk;
  __syncthreads();

  // ---- layer 1: bufB = tanh(bufA @ W1 + b1) ----
  for (int j = 0; j < 16; ++j) {
    for (int e = tid; e < FDIM * 16; e += 128) {       // coop: W1[:, 16j..16j+16]
      const int k = e >> 4, n = e & 15;
      wpan[k * PSTRIDE + n] = pW1[k * FDIM + (j << 4) + n];
    }
    __syncthreads();
    const int ncol = (j << 4) + m;
    v8f acc;
    const float bv = pb1[ncol];
    #pragma unroll
    for (int r = 0; r < 8; ++r) acc[r] = bv;
    acc = gemm_tile_lds(bufA, wpan, lane, acc);
    #pragma unroll
    for (int r = 0; r < 8; ++r)
      bufB[(r + half * 8) * ASTRIDE + ncol] = fast_tanh(acc[r]);
    __syncthreads();
  }

  // ---- layer 2: bufA = tanh(bufB @ W2 + b2)  (X buffer recycled as H2) ----
  for (int j = 0; j < 16; ++j) {
    for (int e = tid; e < FDIM * 16; e += 128) {
      const int k = e >> 4, n = e & 15;
      wpan[k * PSTRIDE + n] = pW2[k * FDIM + (j << 4) + n];
    }
    __syncthreads();
    const int ncol = (j << 4) + m;
    v8f acc;
    const float bv = pb2[ncol];
    #pragma unroll
    for (int r = 0; r < 8; ++r) acc[r] = bv;
    acc = gemm_tile_lds(bufB, wpan, lane, acc);
    #pragma unroll
    for (int r = 0; r < 8; ++r)
      bufA[(r + half * 8) * ASTRIDE + ncol] = fast_tanh(acc[r]);
    __syncthreads();
  }

  // ---- per-atom energy: y = h2 . W3 + b3 + skip + bskip ; scatter to structure ----
  float part = 0.0f;
  for (int f = 0; f < 128; ++f)
    part += bufA[m * ASTRIDE + fbase + f] * pw3[fbase + f];
  part += __shfl_xor(part, 16, 32);
  if (half == 0 && idxm >= 0) {
    float y = part + b3[sp] + auxf[m] + bsk[sp];
    atomicAdd(&energies[smap[idxm]], y);
  }

  // ---- dpre2 = W3 * (1 - h2^2), in place in bufA (wave-local, LDS in-order) ----
  for (int e = lane; e < 16 * FDIM; e += 32) {
    const int row = e >> 8, col = e & (FDIM - 1);
    const float h = bufA[row * ASTRIDE + col];
    bufA[row * ASTRIDE + col] = pw3[col] * (1.0f - h * h);
  }
  __syncthreads();

  // ---- dpre1 = (dpre2 @ W2^T) * (1 - h1^2), in place in bufB ----
  for (int j = 0; j < 16; ++j) {
    for (int e = tid; e < FDIM * 16; e += 128) {       // coop: W2^T panel (coalesced k runs)
      const int n = e >> 8, k = e & 255;
      wpan[k * PSTRIDE + n] = pW2[((j << 4) + n) * FDIM + k];
    }
    __syncthreads();
    const int ncol = (j << 4) + m;
    v8f acc;
    #pragma unroll
    for (int r = 0; r < 8; ++r) acc[r] = 0.0f;
    acc = gemm_tile_lds(bufA, wpan, lane, acc);
    #pragma unroll
    for (int r = 0; r < 8; ++r) {
      const int e = (r + half * 8) * ASTRIDE + ncol;
      const float h1v = bufB[e];
      bufB[e] = acc[r] * (1.0f - h1v * h1v);
    }
    __syncthreads();
  }

  // ---- dx = dpre1 @ W1^T + Wskip  -> nn_grads ----
  for (int j = 0; j < 16; ++j) {
    for (int e = tid; e < FDIM * 16; e += 128) {       // coop: W1^T panel
      const int n = e >> 8, k = e & 255;
      wpan[k * PSTRIDE + n] = pW1[((j << 4) + n) * FDIM + k];
    }
    __syncthreads();
    const int ncol = (j << 4) + m;
    v8f acc;
    const float wv = wsk[ncol];
    #pragma unroll
    for (int r = 0; r < 8; ++r) acc[r] = wv;
    acc = gemm_tile_lds(bufB, wpan, lane, acc);
    #pragma unroll
    for (int r = 0; r < 8; ++r) {
      const int ida = auxi[r + half * 8];
      if (ida >= 0) ngrads[(size_t)ida * FDIM + ncol] = acc[r];
    }
    __syncthreads();
  }
}

// ---------------- forces: wave per pair, streams the 614 MB grad tensor ----------------
__global__ void k_forces(const float* __restrict__ pgrad,  // [P,3,256]
                         const int* __restrict__ gmap,     // [P]
                         const float* __restrict__ ngrads, // [N,256]
                         float* __restrict__ forces,       // [N,3]
                         int P) {
  const int gwave = (int)((blockIdx.x * blockDim.x + threadIdx.x) >> 5);
  const int lane  = threadIdx.x & 31;
  if (gwave >= P) return;
  const int a = gmap[gwave];
  const float* grow = ngrads + (size_t)a * FDIM;
  const float* prow = pgrad + (size_t)gwave * 3 * FDIM;
  const int f0 = lane * 8;
  const float4 g0 = *reinterpret_cast<const float4*>(&grow[f0]);
  const float4 g1 = *reinterpret_cast<const float4*>(&grow[f0 + 4]);
  float sums[3];
  #pragma unroll
  for (int d = 0; d < 3; ++d) {
    const float4 p0 = *reinterpret_cast<const float4*>(&prow[d * FDIM + f0]);
    const float4 p1 = *reinterpret_cast<const float4*>(&prow[d * FDIM + f0 + 4]);
    float s = p0.x * g0.x + p0.y * g0.y + p0.z * g0.z + p0.w * g0.w
            + p1.x * g1.x + p1.y * g1.y + p1.z * g1.z + p1.w * g1.w;
    #pragma unroll
    for (int off = 16; off > 0; off >>= 1) s += __shfl_xor(s, off, 32);
    sums[d] = s;
  }
  if (lane == 0) {
    #pragma unroll
    for (int d = 0; d < 3; ++d)
      atomicAdd(&forces[(size_t)a * 3 + d], -sums[d]);
  }
}

// ---------------- host ----------------
extern "C" void kernel_launch(void* const* d_in, const int* in_sizes, int n_in,
                              void* d_out, int out_size, void* d_ws, size_t ws_size,
                              hipStream_t stream) {
  const float* ps    = (const float*)d_in[0];
  const int*   z     = (const int*)d_in[1];
  const int*   smap  = (const int*)d_in[2];
  const float* pgrad = (const float*)d_in[3];
  const int*   gmap  = (const int*)d_in[4];
  const float* W1  = (const float*)d_in[5];
  const float* b1  = (const float*)d_in[6];
  const float* W2  = (const float*)d_in[7];
  const float* b2  = (const float*)d_in[8];
  const float* W3  = (const float*)d_in[9];
  const float* b3  = (const float*)d_in[10];
  const float* Wsk = (const float*)d_in[11];
  const float* bsk = (const float*)d_in[12];
  const int N = in_sizes[1];   // ps_z has N elements
  const int P = in_sizes[4];   // gradient_map has P elements

  // workspace: counts/cursors[8] | offsets[5] | binned[N+320] | nn_grads[N*256]
  int* counts = (int*)d_ws;
  int* offs   = counts + 8;
  int* binned = counts + 16;
  uintptr_t gaddr = (uintptr_t)(binned + N + 320);
  gaddr = (gaddr + 255) & ~(uintptr_t)255;
  float* ngrads = (float*)gaddr;

  float* energies = (float*)d_out;
  float* forces   = energies + NSTRUCT;

  k_init<<<512, 256, 0, stream>>>(energies, out_size, counts, binned, N + 320);
  k_count<<<(N + 255) / 256, 256, 0, stream>>>(z, counts, N);
  k_offsets<<<1, 1, 0, stream>>>(counts, offs);
  k_scatter<<<(N + 255) / 256, 256, 0, stream>>>(z, counts + 4, offs, binned, N);

  // 64-row blocks; each species bin padded to 64, so <= N + 4*63 rows total
  const int blocks = (N + NSPEC * 63 + 63) / 64;
  const size_t ldsBytes =
      (size_t)(PANEL_FLOATS + WAVES_PER_BLOCK * PERWAVE_FLOATS) * sizeof(float);
  hipFuncSetAttribute((const void*)k_mlp,
                      hipFuncAttributeMaxDynamicSharedMemorySize, (int)ldsBytes);
  k_mlp<<<blocks, WAVES_PER_BLOCK * 32, ldsBytes, stream>>>(
      ps, smap, offs, binned, W1, b1, W2, b2, W3, b3, Wsk, bsk, energies, ngrads);

  const int fblocks = (P + 7) / 8;   // 8 waves / 256-thread block, wave per pair
  k_forces<<<fblocks, 256, 0, stream>>>(pgrad, gmap, ngrads, forces, P);

  (void)n_in; (void)ws_size; (void)out_size;
}